// ClassifierModel_73031623901201
// MI455X (gfx1250) — compile-verified
//
#include <hip/hip_runtime.h>
#include <math.h>

// ---------------------------------------------------------------------------
// Types for CDNA5 WMMA f32 16x16x4:  A=16x4 (2 f32/lane), B=4x16 (2 f32/lane),
// C/D = 16x16 (8 f32/lane).
// ---------------------------------------------------------------------------
typedef __attribute__((ext_vector_type(2))) float    v2f;
typedef __attribute__((ext_vector_type(8))) float    v8f;
typedef __attribute__((ext_vector_type(4))) uint32_t u32x4;
typedef __attribute__((ext_vector_type(8))) uint32_t u32x8;

#define HID   1024
#define EMBD  512
#define SEQ   4096
#define G4H   4096          // 4*HID
#define VOCAB 50257
#define NWG   64            // persistent workgroups for the recurrent scan
#define UPW   16            // hidden units per workgroup (NWG*UPW == HID)
#define WPAD  1028          // row stride in LDS: 1024 + 4 TDM pad DWORDs

// LDS byte address (offset within the wave's LDS allocation) of a pointer.
__device__ __forceinline__ uint32_t lds_addr_of(void* p) {
    return (uint32_t)(uintptr_t)(__attribute__((address_space(3))) void*)p;
}

// ---------------------------------------------------------------------------
// 1) Embedding gather: x[t, :] = emb[input[t], :]
// ---------------------------------------------------------------------------
__global__ __launch_bounds__(256) void embed_gather(
    const int* __restrict__ idx, const float* __restrict__ emb,
    float* __restrict__ x)
{
    const int t   = blockIdx.x;
    const int tok = idx[t];
    const float2* src = (const float2*)(emb + (size_t)tok * EMBD);
    float2*       dst = (float2*)(x + (size_t)t * EMBD);
    dst[threadIdx.x] = src[threadIdx.x];     // 256 threads * 2 = 512 floats
}

// ---------------------------------------------------------------------------
// 2) WMMA f32 GEMM:  Y[M,N] = X[M,K] @ W[N,K]^T + (b0[n] + b1[n])
//    One wave computes a 16x64 strip (A fragment reused over 4 N-subtiles).
//    2-deep register double-buffer: loads for step k+8 issue after the WMMAs
//    of step k, while step k+4 computes from the other buffer -> each load
//    gets half an iteration (4 WMMAs) of latency cover. Requires K % 8 == 0.
// ---------------------------------------------------------------------------
__global__ __launch_bounds__(256) void wmma_gemm_bias(
    const float* __restrict__ X, const float* __restrict__ W,
    const float* __restrict__ b0, const float* __restrict__ b1,
    float* __restrict__ Y, int M, int N, int K)
{
    const int lane  = threadIdx.x & 31;
    const int wave  = threadIdx.x >> 5;
    const int wm    = wave & 3;          // 0..3 (M direction)
    const int wn    = wave >> 2;         // 0..1 (N direction)
    const int row0  = blockIdx.x * 64  + wm * 16;
    const int col0  = blockIdx.y * 128 + wn * 64;
    const int lhalf = lane >> 4;         // 0/1
    const int l16   = lane & 15;

    v8f acc[4] = {};

    // A fragment: lane -> M = row0 + l16; VGPR v -> K = kk + 2*lhalf + v
    const float* aptr = X + (size_t)(row0 + l16) * K + 2 * lhalf;
    const float* bptr[4];
#pragma unroll
    for (int s = 0; s < 4; ++s)
        bptr[s] = W + (size_t)(col0 + s * 16 + l16) * K + 2 * lhalf;

    // Preload pipeline stages for k = 0 and k = 4.
    v2f a_buf[2];
    v2f b_buf[2][4];
    a_buf[0] = *(const v2f*)aptr;
    a_buf[1] = *(const v2f*)(aptr + 4);
#pragma unroll
    for (int s = 0; s < 4; ++s) {
        b_buf[0][s] = *(const v2f*)bptr[s];
        b_buf[1][s] = *(const v2f*)(bptr[s] + 4);
    }

    int kk = 0;
    for (; kk + 8 < K; kk += 8) {
        // ---- compute step kk from buffer 0, then refill it with kk+8 ----
#pragma unroll
        for (int s = 0; s < 4; ++s)
            acc[s] = __builtin_amdgcn_wmma_f32_16x16x4_f32(
                false, a_buf[0], false, b_buf[0][s], (short)0, acc[s],
                false, false);
        a_buf[0] = *(const v2f*)(aptr + kk + 8);
#pragma unroll
        for (int s = 0; s < 4; ++s)
            b_buf[0][s] = *(const v2f*)(bptr[s] + kk + 8);

        // ---- compute step kk+4 from buffer 1, refill it with kk+12 ----
#pragma unroll
        for (int s = 0; s < 4; ++s)
            acc[s] = __builtin_amdgcn_wmma_f32_16x16x4_f32(
                false, a_buf[1], false, b_buf[1][s], (short)0, acc[s],
                false, false);
        a_buf[1] = *(const v2f*)(aptr + kk + 12);
#pragma unroll
        for (int s = 0; s < 4; ++s)
            b_buf[1][s] = *(const v2f*)(bptr[s] + kk + 12);
    }
    // Epilogue: steps K-8 (buffer 0) and K-4 (buffer 1).
#pragma unroll
    for (int s = 0; s < 4; ++s)
        acc[s] = __builtin_amdgcn_wmma_f32_16x16x4_f32(
            false, a_buf[0], false, b_buf[0][s], (short)0, acc[s],
            false, false);
#pragma unroll
    for (int s = 0; s < 4; ++s)
        acc[s] = __builtin_amdgcn_wmma_f32_16x16x4_f32(
            false, a_buf[1], false, b_buf[1][s], (short)0, acc[s],
            false, false);

    // C/D layout: VGPR v -> M = row0 + v + 8*lhalf ; lane -> N = col + l16
#pragma unroll
    for (int s = 0; s < 4; ++s) {
        const int n = col0 + s * 16 + l16;
        const float bias = b0[n] + b1[n];
#pragma unroll
        for (int v = 0; v < 8; ++v) {
            const int m = row0 + v + 8 * lhalf;
            Y[(size_t)m * N + n] = acc[s][v] + bias;
        }
    }
}

// ---------------------------------------------------------------------------
// Agent-scope grid barrier (sense-reversing on a generation counter).
// bar[0] = arrive counter (must start at 0), bar[1] = generation.
// ---------------------------------------------------------------------------
__device__ __forceinline__ void grid_sync(unsigned* bar, unsigned nwg)
{
    __threadfence();
    __syncthreads();
    if (threadIdx.x == 0) {
        unsigned gen = __hip_atomic_load(&bar[1], __ATOMIC_ACQUIRE,
                                         __HIP_MEMORY_SCOPE_AGENT);
        unsigned got = __hip_atomic_fetch_add(&bar[0], 1u, __ATOMIC_ACQ_REL,
                                              __HIP_MEMORY_SCOPE_AGENT);
        if (got == nwg - 1u) {
            __hip_atomic_store(&bar[0], 0u, __ATOMIC_RELAXED,
                               __HIP_MEMORY_SCOPE_AGENT);
            __hip_atomic_fetch_add(&bar[1], 1u, __ATOMIC_RELEASE,
                                   __HIP_MEMORY_SCOPE_AGENT);
        } else {
            while (__hip_atomic_load(&bar[1], __ATOMIC_ACQUIRE,
                                     __HIP_MEMORY_SCOPE_AGENT) == gen)
                __builtin_amdgcn_s_sleep(1);
        }
    }
    __syncthreads();
}

__device__ __forceinline__ float sigm(float v) { return 1.f / (1.f + expf(-v)); }

// ---------------------------------------------------------------------------
// 3) Persistent LSTM scan. 64 WGs, each owns 16 hidden units (= 64 rows of
//    w_hh). The 4 gate tiles (16 rows x 1024 f32) are DMA'd into LDS by the
//    Tensor Data Mover with 1-DWORD-per-256 padding (row stride 1028 ->
//    conflict-free banking). h ping-pongs through a 2xH global buffer and is
//    staged into LDS each step with the async global->LDS copy path.
// ---------------------------------------------------------------------------
__global__ __launch_bounds__(256) void lstm_scan(
    const float* __restrict__ xg,     // [T, 4H] precomputed x@W_ih.T + biases
    const float* __restrict__ w_hh,   // [4H, H]
    const float* __restrict__ h0,     // [H]
    const float* __restrict__ c0,     // [H]
    float* __restrict__ hs,           // [T, H] hidden sequence out
    float* __restrict__ hbuf,         // [2, H] ping-pong h buffer
    float* __restrict__ hn_out,       // [H] final h
    float* __restrict__ cn_out,       // [H] final c
    unsigned* __restrict__ bar, int T)
{
    extern __shared__ float smem[];
    float* Wl = smem;                         // 64 rows * WPAD (TDM padded)
    float* hl = Wl + 64 * WPAD;               // 1024
    float* gl = hl + HID;                     // 64 gate pre-activations
    float* cl = gl + 64;                      // 16 cell states

    const int tid   = threadIdx.x;
    const int ubase = blockIdx.x * UPW;

    // ---- TDM stage: one tensor_load_to_lds per gate tile (wave 0 only;
    //      TDM ignores EXEC, so one issue per wave == one DMA). ----
    if ((tid >> 5) == 0) {
        const uint32_t lds0 = lds_addr_of(Wl);
#pragma unroll
        for (int g = 0; g < 4; ++g) {
            const uint64_t ga =
                (uint64_t)(uintptr_t)(w_hh + (size_t)(g * HID + ubase) * HID);
            u32x4 d0;                          // D# group 0
            d0.x = 1u;                         // count=1, user mode
            d0.y = lds0 + (uint32_t)(g * 16 * WPAD * 4);   // lds_addr (bytes)
            d0.z = (uint32_t)ga;                           // global_addr[31:0]
            d0.w = (uint32_t)((ga >> 32) & 0x01FFFFFFu)    // global_addr[56:32]
                 | (2u << 30);                             // type = 2 (image)
            u32x8 d1;                          // D# group 1
            d1.s0 = (2u << 16)                 // data_size = 4B
                  | (1u << 20)                 // pad_enable
                  | (7u << 22);                // pad_interval: every 256 DW
                                               // pad_amount = 0 -> 1 DWORD
            d1.s1 = (uint32_t)HID << 16;       // tensor_dim0 = 1024
            d1.s2 = (uint32_t)G4H << 16;       // tensor_dim1 = 4096
            d1.s3 = (uint32_t)HID << 16;       // tile_dim0   = 1024
            d1.s4 = 16u;                       // tile_dim1   = 16 rows
            d1.s5 = (uint32_t)HID;             // tensor_dim0_stride = 1024
            d1.s6 = 0u;
            d1.s7 = 0u;
            asm volatile("tensor_load_to_lds %0, %1"
                         :: "s"(d0), "s"(d1) : "memory");
        }
        __builtin_amdgcn_s_wait_tensorcnt(0);
    }
    if (tid < UPW) {
        hbuf[ubase + tid] = h0[ubase + tid];  // init ping buffer slice
        cl[tid]           = c0[ubase + tid];
    }
    grid_sync(bar, gridDim.x);                // W in LDS + h_0 visible

    const int rlocal = tid >> 2;              // 0..63 : local gate row
    const int klane  = tid & 3;               // 0..3  : K-splitter
    const float* wrow = Wl + rlocal * WPAD;
    const uint32_t hl_lds = lds_addr_of(hl) + (uint32_t)(tid * 16);

    for (int t = 0; t < T; ++t) {
        // Async copy h_{t-1} (4KB) global -> LDS, no VGPR bounce.
        {
            const uint64_t ga = (uint64_t)(uintptr_t)(hbuf + (t & 1) * HID)
                              + (uint32_t)(tid * 16);
            asm volatile("global_load_async_to_lds_b128 %0, %1, off"
                         :: "v"(hl_lds), "v"(ga) : "memory");
            asm volatile("s_wait_asynccnt 0x0" ::: "memory");
        }
        __syncthreads();

        // 4 threads per row; element k sits at LDS offset k + (k>>8) due to
        // TDM padding, so walk 4 blocks of 64 with base 257*b + klane.
        float acc = 0.f;
#pragma unroll
        for (int b = 0; b < 4; ++b) {
            const float* wseg = wrow + 257 * b + klane;
            const float* hseg = hl   + 256 * b + klane;
#pragma unroll 16
            for (int j = 0; j < 64; ++j)
                acc = fmaf(wseg[4 * j], hseg[4 * j], acc);
        }
        acc += __shfl_xor(acc, 1, 32);
        acc += __shfl_xor(acc, 2, 32);
        if (klane == 0) {
            const int gate = rlocal >> 4, u = rlocal & 15;
            gl[rlocal] = acc + xg[(size_t)t * G4H + gate * HID + ubase + u];
        }
        __syncthreads();

        if (tid < UPW) {
            const float i_ = sigm(gl[0 * 16 + tid]);
            const float f_ = sigm(gl[1 * 16 + tid]);
            const float g_ = tanhf(gl[2 * 16 + tid]);
            const float o_ = sigm(gl[3 * 16 + tid]);
            const float cn = f_ * cl[tid] + i_ * g_;
            cl[tid] = cn;
            const float hn = o_ * tanhf(cn);
            hs[(size_t)t * HID + ubase + tid]       = hn;
            hbuf[((t + 1) & 1) * HID + ubase + tid] = hn;  // pong buffer
            if (t == T - 1) {
                hn_out[ubase + tid] = hn;
                cn_out[ubase + tid] = cn;
            }
        }
        grid_sync(bar, gridDim.x);
    }
}

// ---------------------------------------------------------------------------
// 4) Head GEMV: logits[v] = dot(h_last, w_out[v]) + b_out[v]. One wave/row.
// ---------------------------------------------------------------------------
__global__ __launch_bounds__(256) void logits_gemv(
    const float* __restrict__ h, const float* __restrict__ Wout,
    const float* __restrict__ bout, float* __restrict__ logits, int V)
{
    const int row = blockIdx.x * 8 + (threadIdx.x >> 5);
    if (row >= V) return;
    const int lane = threadIdx.x & 31;
    const float* w = Wout + (size_t)row * HID;
    float acc = 0.f;
    for (int k = lane * 4; k < HID; k += 32 * 4) {
        float4 wv = *(const float4*)(w + k);
        float4 hv = *(const float4*)(h + k);
        acc = fmaf(wv.x, hv.x, acc);
        acc = fmaf(wv.y, hv.y, acc);
        acc = fmaf(wv.z, hv.z, acc);
        acc = fmaf(wv.w, hv.w, acc);
    }
#pragma unroll
    for (int off = 16; off; off >>= 1) acc += __shfl_xor(acc, off, 32);
    if (lane == 0) logits[row] = acc + bout[row];
}

// 5) Single-block two-pass log-sum-exp.
__global__ __launch_bounds__(1024) void lse_reduce(
    const float* __restrict__ logits, float* __restrict__ lse, int V)
{
    __shared__ float red[32];
    const int tid = threadIdx.x, lane = tid & 31, wv = tid >> 5;
    float m = -INFINITY;
    for (int i = tid; i < V; i += 1024) m = fmaxf(m, logits[i]);
#pragma unroll
    for (int off = 16; off; off >>= 1) m = fmaxf(m, __shfl_xor(m, off, 32));
    if (lane == 0) red[wv] = m;
    __syncthreads();
    if (tid < 32) {
        float v = red[tid];
#pragma unroll
        for (int off = 16; off; off >>= 1) v = fmaxf(v, __shfl_xor(v, off, 32));
        if (tid == 0) red[0] = v;
    }
    __syncthreads();
    m = red[0];
    __syncthreads();
    float s = 0.f;
    for (int i = tid; i < V; i += 1024) s += expf(logits[i] - m);
#pragma unroll
    for (int off = 16; off; off >>= 1) s += __shfl_xor(s, off, 32);
    if (lane == 0) red[wv] = s;
    __syncthreads();
    if (tid == 0) {
        float tot = 0.f;
        for (int w = 0; w < 32; ++w) tot += red[w];
        *lse = m + logf(tot);
    }
}

__global__ __launch_bounds__(256) void logsoftmax_finalize(
    const float* __restrict__ logits, const float* __restrict__ lse,
    float* __restrict__ out, int V)
{
    const int i = blockIdx.x * 256 + threadIdx.x;
    if (i < V) out[i] = logits[i] - *lse;
}

// ---------------------------------------------------------------------------
// Host launcher
// ---------------------------------------------------------------------------
extern "C" void kernel_launch(void* const* d_in, const int* in_sizes, int n_in,
                              void* d_out, int out_size, void* d_ws, size_t ws_size,
                              hipStream_t stream)
{
    (void)in_sizes; (void)n_in; (void)out_size; (void)ws_size;
    const int*   input = (const int*)  d_in[0];
    const float* h0    = (const float*)d_in[1];   // [2,1,H]
    const float* c0    = (const float*)d_in[2];
    const float* emb   = (const float*)d_in[3];
    const float* w_ih0 = (const float*)d_in[4];
    const float* w_hh0 = (const float*)d_in[5];
    const float* b_ih0 = (const float*)d_in[6];
    const float* b_hh0 = (const float*)d_in[7];
    const float* w_ih1 = (const float*)d_in[8];
    const float* w_hh1 = (const float*)d_in[9];
    const float* b_ih1 = (const float*)d_in[10];
    const float* b_hh1 = (const float*)d_in[11];
    const float* w_out = (const float*)d_in[12];
    const float* b_out = (const float*)d_in[13];
    float* out = (float*)d_out;

    // Workspace carve-up (floats).
    float* ws     = (float*)d_ws;
    float* x      = ws;                                   // SEQ*EMBD
    float* xg     = x   + (size_t)SEQ * EMBD;             // SEQ*4H (reused)
    float* hs0    = xg  + (size_t)SEQ * G4H;              // SEQ*H
    float* hs1    = hs0 + (size_t)SEQ * HID;              // SEQ*H
    float* hbuf   = hs1 + (size_t)SEQ * HID;              // 2*H
    float* logits = hbuf + 2 * HID;                       // VOCAB (rounded)
    float* lse    = logits + 50272;
    unsigned* bar = (unsigned*)(lse + 4);

    hipMemsetAsync(bar, 0, 16, stream);                   // arrive ctr must be 0

    // d_out layout: log_softmax [50257] | h_n [2*H] | c_n [2*H]
    float* out_lp = out;
    float* hn0 = out + VOCAB;
    float* hn1 = hn0 + HID;
    float* cn0 = hn1 + HID;
    float* cn1 = cn0 + HID;

    embed_gather<<<SEQ, 256, 0, stream>>>(input, emb, x);

    dim3 g(SEQ / 64, G4H / 128);
    wmma_gemm_bias<<<g, 256, 0, stream>>>(x, w_ih0, b_ih0, b_hh0, xg,
                                          SEQ, G4H, EMBD);

    const size_t smem = (size_t)(64 * WPAD + HID + 64 + 16) * sizeof(float);
    hipFuncSetAttribute((const void*)lstm_scan,
                        hipFuncAttributeMaxDynamicSharedMemorySize, (int)smem);

    lstm_scan<<<NWG, 256, smem, stream>>>(xg, w_hh0, h0, c0, hs0, hbuf,
                                          hn0, cn0, bar, SEQ);

    wmma_gemm_bias<<<g, 256, 0, stream>>>(hs0, w_ih1, b_ih1, b_hh1, xg,
                                          SEQ, G4H, HID);

    lstm_scan<<<NWG, 256, smem, stream>>>(xg, w_hh1, h0 + HID, c0 + HID, hs1,
                                          hbuf, hn1, cn1, bar, SEQ);

    const float* hlast = hs1 + (size_t)(SEQ - 1) * HID;
    logits_gemv<<<(VOCAB + 7) / 8, 256, 0, stream>>>(hlast, w_out, b_out,
                                                     logits, VOCAB);
    lse_reduce<<<1, 1024, 0, stream>>>(logits, lse, VOCAB);
    logsoftmax_finalize<<<(VOCAB + 255) / 256, 256, 0, stream>>>(logits, lse,
                                                                 out_lp, VOCAB);
}